// ResidualVectorQuantizer_51608327028990
// MI455X (gfx1250) — compile-verified
//
#include <hip/hip_runtime.h>
#include <hip/hip_bf16.h>

// Problem constants (from reference setup_inputs)
#define RB 16
#define RD 512
#define RF 1500
#define RQ 8
#define RC 1024
#define RN (RB * RF)           // 24000 rows
#define NTILES (RN / 16)       // 1500 row tiles
#define PITCH 514              // LDS row pitch (floats), padded vs 512 to dodge bank conflicts

#define OUT_QUANT_OFF  ((size_t)0)
#define OUT_CODES_OFF  ((size_t)RB * RD * RF)                  // 12,288,000
#define OUT_COMMIT_OFF (OUT_CODES_OFF + (size_t)RB * RQ * RF)  // 12,480,000
#define OUT_CBLOSS_OFF (OUT_COMMIT_OFF + 1)

typedef __attribute__((ext_vector_type(8))) float v8f;
typedef __attribute__((ext_vector_type(2))) float v2f;

// ---------------------------------------------------------------------------
// Kernel 1: half squared norms of every codebook row, and zero loss accums.
// ---------------------------------------------------------------------------
__global__ void rvq_halfnorm(const float* __restrict__ cb,
                             float* __restrict__ hn,
                             float* __restrict__ lossAcc) {
  int c = blockIdx.x * blockDim.x + threadIdx.x;   // 0 .. RQ*RC-1
  if (c < RQ * RC) {
    const float* row = cb + (size_t)c * RD;
    float s = 0.f;
    for (int d = 0; d < RD; ++d) { float v = row[d]; s += v * v; }
    hn[c] = 0.5f * s;
  }
  if (c < RQ) lossAcc[c] = 0.f;
}

// ---------------------------------------------------------------------------
// Kernel 2: fused 8-stage RVQ for one 16-row tile, residual lives in LDS.
// Distance GEMM on V_WMMA_F32_16X16X4_F32 (f32 matrix pipe, exact precision).
// ---------------------------------------------------------------------------
__global__ __launch_bounds__(256) void rvq_main(const float* __restrict__ latents,
                                                const float* __restrict__ codebooks,
                                                const float* __restrict__ halfnorm,
                                                float* __restrict__ out,
                                                float* __restrict__ lossAcc) {
  __shared__ float A[16 * PITCH];       // residual tile [16 rows][512 d], padded
  __shared__ float redVal[8 * 16];      // per-wave best score per row
  __shared__ int   redIdx[8 * 16];
  __shared__ int   idxFinal[16];

  const int tile = blockIdx.x;          // 0..1499
  const int tid  = threadIdx.x;         // 0..255
  const int lane = tid & 31;
  const int wave = tid >> 5;            // 0..7

  // ---- Load residual tile: element (m,d) sits at latents[b*D*F + d*F + f],
  //      n = tile*16 + m, b = n/F, f = n%F.  m = tid%16 -> consecutive f ->
  //      consecutive addresses within a d-slice (coalesced 64B groups).
  {
    const int m  = tid & 15;
    const int d0 = tid >> 4;                 // 0..15
    const int n  = tile * 16 + m;
    const int b  = n / RF, f = n % RF;
    const float* src = latents + (size_t)b * RD * RF + f;
    #pragma unroll
    for (int i = 0; i < 32; ++i) {
      const int d = d0 + 16 * i;
      A[m * PITCH + d] = src[(size_t)d * RF];
    }
  }
  __syncthreads();

  // WMMA f32 16x16x4 per-lane layout (ISA 7.12.2, 32-bit A 16x4):
  //   lane L : A row m = L%16, K pair = 2*(L>=16) + {0,1}
  //   C/D    : VGPR r -> row r + 8*(L>=16), col = ctile*16 + L%16
  const int mA    = lane & 15;
  const int khalf = lane >> 4;

  for (int q = 0; q < RQ; ++q) {
    const float* cb = codebooks + (size_t)q * RC * RD;
    const float* hn = halfnorm + q * RC;

    float bestVal[8];
    int   bestIdx[8];
    #pragma unroll
    for (int r = 0; r < 8; ++r) { bestVal[r] = -1e30f; bestIdx[r] = 0; }

    // Each wave owns 8 column tiles (16 codes each) -> all 1024 codes covered.
    for (int j = 0; j < 8; ++j) {
      const int ctile = wave * 8 + j;
      const int c     = ctile * 16 + (lane & 15);
      const float* bsrc = cb + (size_t)c * RD + 2 * khalf;

      v8f acc = {};
      #pragma unroll 8
      for (int k = 0; k < RD; k += 4) {
        v2f av, bv;
        av[0] = A[mA * PITCH + k + 2 * khalf];
        av[1] = A[mA * PITCH + k + 2 * khalf + 1];
        bv[0] = bsrc[k];
        bv[1] = bsrc[k + 1];
        acc = __builtin_amdgcn_wmma_f32_16x16x4_f32(
            /*neg_a=*/false, av, /*neg_b=*/false, bv,
            /*c_mod=*/(short)0, acc, /*reuse_a=*/false, /*reuse_b=*/false);
      }

      // score = r.c - 0.5*||c||^2 ; argmin(dist) == argmax(score)
      const float h = hn[c];
      #pragma unroll
      for (int r = 0; r < 8; ++r) {
        const float s = acc[r] - h;
        if (s > bestVal[r]) { bestVal[r] = s; bestIdx[r] = c; }
      }
    }

    // Reduce across the 16 lanes of each half (same row, different codes).
    #pragma unroll
    for (int r = 0; r < 8; ++r) {
      float v = bestVal[r]; int id = bestIdx[r];
      #pragma unroll
      for (int off = 1; off < 16; off <<= 1) {
        const float ov = __shfl_xor(v, off, 32);
        const int   oi = __shfl_xor(id, off, 32);
        if (ov > v || (ov == v && oi < id)) { v = ov; id = oi; }  // tie -> low idx
      }
      bestVal[r] = v; bestIdx[r] = id;
    }
    if ((lane & 15) == 0) {
      #pragma unroll
      for (int r = 0; r < 8; ++r) {
        const int row = r + 8 * khalf;
        redVal[wave * 16 + row] = bestVal[r];
        redIdx[wave * 16 + row] = bestIdx[r];
      }
    }
    __syncthreads();

    // Reduce across the 8 waves; write codes for this stage.
    if (tid < 16) {
      float v = redVal[tid]; int id = redIdx[tid];
      for (int w = 1; w < 8; ++w) {
        const float ov = redVal[w * 16 + tid];
        const int   oi = redIdx[w * 16 + tid];
        if (ov > v || (ov == v && oi < id)) { v = ov; id = oi; }
      }
      idxFinal[tid] = id;
      const int n = tile * 16 + tid;
      const int b = n / RF, f = n % RF;
      out[OUT_CODES_OFF + (size_t)b * RQ * RF + (size_t)q * RF + f] = (float)id;
    }
    __syncthreads();

    // Residual update in LDS + loss (sum of squared new residual).
    {
      const int m  = tid >> 4;     // 0..15, wave covers 2 rows: coalesced cb reads
      const int d0 = tid & 15;
      const int cs = idxFinal[m];
      const float* crow = cb + (size_t)cs * RD;
      float ss = 0.f;
      #pragma unroll
      for (int i = 0; i < 32; ++i) {
        const int d = d0 + 16 * i;
        const float a = A[m * PITCH + d] - crow[d];
        A[m * PITCH + d] = a;
        ss += a * a;
      }
      #pragma unroll
      for (int off = 1; off < 32; off <<= 1) ss += __shfl_xor(ss, off, 32);
      if (lane == 0) atomicAdd(&lossAcc[q], ss);
    }
    __syncthreads();
  }

  // quantized = latents - final residual, back to [B, D, F] layout.
  {
    const int m  = tid & 15;
    const int d0 = tid >> 4;
    const int n  = tile * 16 + m;
    const int b  = n / RF, f = n % RF;
    const float* src = latents + (size_t)b * RD * RF + f;
    float*       dst = out + OUT_QUANT_OFF + (size_t)b * RD * RF + f;
    #pragma unroll
    for (int i = 0; i < 32; ++i) {
      const int d = d0 + 16 * i;
      dst[(size_t)d * RF] = src[(size_t)d * RF] - A[m * PITCH + d];
    }
  }
}

// ---------------------------------------------------------------------------
// Kernel 3: finalize the two loss scalars (forward values are identical).
// ---------------------------------------------------------------------------
__global__ void rvq_finalize(const float* __restrict__ lossAcc,
                             float* __restrict__ out) {
  if (threadIdx.x == 0 && blockIdx.x == 0) {
    float s = 0.f;
    for (int q = 0; q < RQ; ++q) s += lossAcc[q];
    const float loss = s / (float)RQ / (float)((size_t)RB * RD * RF);
    out[OUT_COMMIT_OFF] = loss;
    out[OUT_CBLOSS_OFF] = loss;
  }
}

extern "C" void kernel_launch(void* const* d_in, const int* in_sizes, int n_in,
                              void* d_out, int out_size, void* d_ws, size_t ws_size,
                              hipStream_t stream) {
  const float* latents   = (const float*)d_in[0];  // [16, 512, 1500] f32
  const float* codebooks = (const float*)d_in[1];  // [8, 1024, 512] f32
  float* out = (float*)d_out;

  float* hn      = (float*)d_ws;          // 8192 floats
  float* lossAcc = hn + RQ * RC;          // 8 floats

  rvq_halfnorm<<<(RQ * RC + 255) / 256, 256, 0, stream>>>(codebooks, hn, lossAcc);
  rvq_main<<<NTILES, 256, 0, stream>>>(latents, codebooks, hn, out, lossAcc);
  rvq_finalize<<<1, 32, 0, stream>>>(lossAcc, out);
}